// Encoder_18116172055321
// MI455X (gfx1250) — compile-verified
//
#include <hip/hip_runtime.h>
#include <hip/hip_bf16.h>
#include <math.h>

#define B_   256
#define T_   256
#define EMB_ 128
#define IND  384       // 3*EMB
#define UN   100       // UNITS
#define G4   400       // 4*UNITS

typedef __attribute__((ext_vector_type(2))) float v2f;
typedef __attribute__((ext_vector_type(8))) float v8f;

__device__ __forceinline__ float sigf(float x) { return 1.0f / (1.0f + expf(-x)); }

// ---------------------------------------------------------------------------
// Kernel 1: fused embedding-gather + input GEMM
//   XW[m, n] = concat(E_word[words[m]], E_pos[pos[m]], E_dep[dep[m]]) . W[:, n] + b[n]
// M = B*T rows, N = 800 (fwd cols 0..399 | bwd cols 0..399), K = 384.
// One workgroup = 16 rows x 800 cols; A tile staged in LDS; WMMA f32 16x16x4.
// ---------------------------------------------------------------------------
__global__ __launch_bounds__(256) void xw_gemm_kernel(
    const int* __restrict__ words, const int* __restrict__ pos, const int* __restrict__ dep,
    const float* __restrict__ Ew, const float* __restrict__ Ep, const float* __restrict__ Ed,
    const float* __restrict__ Wf, const float* __restrict__ Wb,
    const float* __restrict__ bf, const float* __restrict__ bb,
    float* __restrict__ XWf, float* __restrict__ XWb)
{
    __shared__ float At[16 * 385];           // stride 385 -> conflict-free column reads
    const int tid  = threadIdx.x;
    const int row0 = blockIdx.x * 16;

    // Gather 16x384 A tile from the three embedding tables.
    for (int i = tid; i < 16 * IND; i += 256) {
        const int r = i / IND;
        const int k = i - r * IND;
        const int m = row0 + r;
        float v;
        if (k < EMB_)          v = Ew[(size_t)words[m] * EMB_ + k];
        else if (k < 2 * EMB_) v = Ep[(size_t)pos[m]   * EMB_ + (k - EMB_)];
        else                   v = Ed[(size_t)dep[m]   * EMB_ + (k - 2 * EMB_)];
        At[r * 385 + k] = v;
    }
    __syncthreads();

    const int wid  = tid >> 5;
    const int lane = tid & 31;
    const int l    = lane & 15;
    const int hi   = lane >> 4;

    // 50 column tiles of 16: tiles 0..24 -> fwd, 25..49 -> bwd.
    for (int tile = wid; tile < 50; tile += 8) {
        const bool  bwd = (tile >= 25);
        const int   nn  = (tile - (bwd ? 25 : 0)) * 16 + l;
        const float* W  = bwd ? Wb : Wf;
        const float bias = bwd ? bb[nn] : bf[nn];

        v8f c;
        #pragma unroll
        for (int j = 0; j < 8; ++j) c[j] = bias;

        #pragma unroll 4
        for (int k = 0; k < IND; k += 4) {
            const int ka = k + 2 * hi;
            v2f a, b;
            a[0] = At[l * 385 + ka];
            a[1] = At[l * 385 + ka + 1];
            b[0] = W[(size_t)ka * G4 + nn];
            b[1] = W[(size_t)(ka + 1) * G4 + nn];
            c = __builtin_amdgcn_wmma_f32_16x16x4_f32(false, a, false, b,
                                                      (short)0, c, false, false);
        }

        float* XW = bwd ? XWb : XWf;
        #pragma unroll
        for (int j = 0; j < 8; ++j) {
            const int row = row0 + j + 8 * hi;
            XW[(size_t)row * G4 + nn] = c[j];
        }
    }
}

// ---------------------------------------------------------------------------
// Kernel 2: masked LSTM recurrence, batch-parallel.
// grid = 32 workgroups: blockIdx 0..15 = forward batch tiles, 16..31 = backward.
// Per step: z[16,400] = XW[., t, .] + h[16,100] @ U[100,400] via WMMA (U from L2),
// then fused gate math + Keras mask carry-through in LDS.
// ---------------------------------------------------------------------------
__global__ __launch_bounds__(256) void lstm_kernel(
    const float* __restrict__ XWf, const float* __restrict__ XWb,
    const float* __restrict__ Uf,  const float* __restrict__ Ub,
    const int*   __restrict__ words,
    float* __restrict__ Hfo, float* __restrict__ Hbo)
{
    __shared__ float h_s[16 * 101];   // stride 101 -> conflict-free column reads
    __shared__ float c_s[16 * 101];
    __shared__ float z_s[16 * G4];

    const int dir = blockIdx.x >> 4;            // 0 = fwd, 1 = bwd
    const int b0  = (blockIdx.x & 15) * 16;
    const float* XW = dir ? XWb : XWf;
    const float* U  = dir ? Ub  : Uf;
    float*       H  = dir ? Hbo : Hfo;

    const int tid = threadIdx.x;
    for (int i = tid; i < 16 * 101; i += 256) { h_s[i] = 0.0f; c_s[i] = 0.0f; }
    __syncthreads();

    const int wid  = tid >> 5;
    const int lane = tid & 31;
    const int l    = lane & 15;
    const int hi   = lane >> 4;

    for (int step = 0; step < T_; ++step) {
        const int t = dir ? (T_ - 1 - step) : step;

        // z = xw_t + h @ U   (25 column tiles of 16, K = 100 -> 25 wmma k-steps)
        for (int tile = wid; tile < 25; tile += 8) {
            const int n = tile * 16 + l;
            v8f c;
            #pragma unroll
            for (int j = 0; j < 8; ++j) {
                const int r = j + 8 * hi;
                c[j] = XW[((size_t)(b0 + r) * T_ + t) * G4 + n];
            }
            #pragma unroll 5
            for (int k = 0; k < UN; k += 4) {
                const int ka = k + 2 * hi;
                v2f a, b;
                a[0] = h_s[l * 101 + ka];
                a[1] = h_s[l * 101 + ka + 1];
                b[0] = U[(size_t)ka * G4 + n];
                b[1] = U[(size_t)(ka + 1) * G4 + n];
                c = __builtin_amdgcn_wmma_f32_16x16x4_f32(false, a, false, b,
                                                          (short)0, c, false, false);
            }
            #pragma unroll
            for (int j = 0; j < 8; ++j) {
                const int r = j + 8 * hi;
                z_s[r * G4 + n] = c[j];
            }
        }
        __syncthreads();

        // Fused gates: i,f,g,o at columns u, u+100, u+200, u+300.
        for (int i = tid; i < 16 * UN; i += 256) {
            const int r = i / UN;
            const int u = i - r * UN;
            const float zi = z_s[r * G4 + u];
            const float zf = z_s[r * G4 + u + UN];
            const float zg = z_s[r * G4 + u + 2 * UN];
            const float zo = z_s[r * G4 + u + 3 * UN];
            const float ig = sigf(zi);
            const float fg = sigf(zf);
            const float gg = tanhf(zg);
            const float og = sigf(zo);
            const float cprev = c_s[r * 101 + u];
            const float cn = fg * cprev + ig * gg;
            const float hn = og * tanhf(cn);
            const bool  msk = words[(size_t)(b0 + r) * T_ + t] != 0;
            const float hprev = h_s[r * 101 + u];
            const float h2 = msk ? hn : hprev;   // carry-through on masked steps
            const float c2 = msk ? cn : cprev;
            h_s[r * 101 + u] = h2;
            c_s[r * 101 + u] = c2;
            H[((size_t)(b0 + r) * T_ + t) * UN + u] = h2;
        }
        __syncthreads();
    }
}

// ---------------------------------------------------------------------------
// Kernel 3: out[b,t] = sigmoid([h_f ; h_b] . W_out + b_out)
// ---------------------------------------------------------------------------
__global__ __launch_bounds__(256) void out_kernel(
    const float* __restrict__ Hf, const float* __restrict__ Hb,
    const float* __restrict__ Wout, const float* __restrict__ bout,
    float* __restrict__ out)
{
    __shared__ float w[2 * UN];
    const int tid = threadIdx.x;
    if (tid < 2 * UN) w[tid] = Wout[tid];
    __syncthreads();

    const int bt = blockIdx.x * 256 + tid;
    const float* hf = Hf + (size_t)bt * UN;
    const float* hb = Hb + (size_t)bt * UN;
    float acc = bout[0];
    #pragma unroll 4
    for (int u = 0; u < UN; ++u) acc += hf[u] * w[u] + hb[u] * w[UN + u];
    out[bt] = sigf(acc);
}

// ---------------------------------------------------------------------------
extern "C" void kernel_launch(void* const* d_in, const int* in_sizes, int n_in,
                              void* d_out, int out_size, void* d_ws, size_t ws_size,
                              hipStream_t stream) {
    (void)in_sizes; (void)n_in; (void)out_size; (void)ws_size;
    const int*   words = (const int*)d_in[0];
    const int*   pos   = (const int*)d_in[1];
    const int*   dep   = (const int*)d_in[2];
    const float* Ew    = (const float*)d_in[3];
    const float* Ep    = (const float*)d_in[4];
    const float* Ed    = (const float*)d_in[5];
    const float* Wf    = (const float*)d_in[6];
    const float* Uf    = (const float*)d_in[7];
    const float* bf    = (const float*)d_in[8];
    const float* Wb    = (const float*)d_in[9];
    const float* Ub    = (const float*)d_in[10];
    const float* bb    = (const float*)d_in[11];
    const float* Wout  = (const float*)d_in[12];
    const float* bout  = (const float*)d_in[13];
    float* outp = (float*)d_out;

    // Workspace layout (floats): XWf | XWb | Hf | Hb  (~262 MB total)
    float* XWf = (float*)d_ws;
    float* XWb = XWf + (size_t)B_ * T_ * G4;
    float* Hf  = XWb + (size_t)B_ * T_ * G4;
    float* Hb  = Hf  + (size_t)B_ * T_ * UN;

    xw_gemm_kernel<<<(B_ * T_) / 16, 256, 0, stream>>>(
        words, pos, dep, Ew, Ep, Ed, Wf, Wb, bf, bb, XWf, XWb);
    lstm_kernel<<<32, 256, 0, stream>>>(XWf, XWb, Uf, Ub, words, Hf, Hb);
    out_kernel<<<(B_ * T_) / 256, 256, 0, stream>>>(Hf, Hb, Wout, bout, outp);
}